// AdaptiveVectorQuantizer_20246475833508
// MI455X (gfx1250) — compile-verified
//
#include <hip/hip_runtime.h>
#include <hip/hip_bf16.h>

typedef float v2f __attribute__((ext_vector_type(2)));
typedef float v8f __attribute__((ext_vector_type(8)));

#define DDIM 512
#define PCB  1024
#define NROW 16384   // B*T = 16*1024
#define NLEV 10
#define GATHER_BLOCKS 2048

// ---------------- Threefry-2x32 (matches JAX) ----------------
__device__ __forceinline__ unsigned rotl32(unsigned x, int r) {
    return (x << r) | (x >> (32 - r));
}

__device__ __forceinline__ void tf2x32(unsigned k0, unsigned k1,
                                       unsigned& x0, unsigned& x1) {
    unsigned ks2 = 0x1BD11BDAu ^ k0 ^ k1;
    x0 += k0; x1 += k1;
#define TF_GRP(a, b, c, d)                                        \
    x0 += x1; x1 = rotl32(x1, a); x1 ^= x0;                       \
    x0 += x1; x1 = rotl32(x1, b); x1 ^= x0;                       \
    x0 += x1; x1 = rotl32(x1, c); x1 ^= x0;                       \
    x0 += x1; x1 = rotl32(x1, d); x1 ^= x0;
    TF_GRP(13, 15, 26, 6)   x0 += k1;  x1 += ks2 + 1u;
    TF_GRP(17, 29, 16, 24)  x0 += ks2; x1 += k0 + 2u;
    TF_GRP(13, 15, 26, 6)   x0 += k0;  x1 += k1 + 3u;
    TF_GRP(17, 29, 16, 24)  x0 += k1;  x1 += ks2 + 4u;
    TF_GRP(13, 15, 26, 6)   x0 += ks2; x1 += k0 + 5u;
#undef TF_GRP
}

__device__ __forceinline__ int make_src(unsigned bits, int j, float cp) {
    unsigned fb = (bits >> 9) | 0x3F800000u;
    float u = __uint_as_float(fb) - 1.0f;           // [0,1)
    if (u <= cp) return j;
    float step = (1.0f - cp) / 1024.0f;
    int off = (int)floorf((u - cp) / step);
    return (j + 1 + off) & (PCB - 1);
}

__global__ __launch_bounds__(512) void rng_src_kernel(
        const float* __restrict__ cp_ptr, int na, int* __restrict__ src) {
    int j = threadIdx.x;                 // 0..511
    // folded key = threefry([0,42], [0,na])
    unsigned kx0 = 0u, kx1 = (unsigned)na;
    tf2x32(0u, 42u, kx0, kx1);
    unsigned x0 = (unsigned)j, x1 = (unsigned)(j + 512);
    tf2x32(kx0, kx1, x0, x1);
    float cp = *cp_ptr;
    src[j]       = make_src(x0, j,       cp);
    src[j + 512] = make_src(x1, j + 512, cp);
}

// ---------------- column norms ‖c_j‖² (level-invariant) ----------------
__global__ __launch_bounds__(256) void colnorm_kernel(
        const float* __restrict__ cb, float* __restrict__ cn2) {
    __shared__ float red[256];
    int r = blockIdx.x, t = threadIdx.x;
    float a = cb[(size_t)r * DDIM + t];
    float b = cb[(size_t)r * DDIM + t + 256];
    red[t] = a * a + b * b;
    __syncthreads();
    for (int off = 128; off > 0; off >>= 1) {
        if (t < off) red[t] += red[t + off];
        __syncthreads();
    }
    if (t == 0) cn2[r] = red[0];
}

// ---------------- deterministic inverse scatter ----------------
__global__ __launch_bounds__(256) void build_lists_kernel(
        const int* __restrict__ src, int* __restrict__ cnt, int* __restrict__ lst) {
    __shared__ int ssrc[PCB];
    for (int i = threadIdx.x; i < PCB; i += 256) ssrc[i] = src[i];
    __syncthreads();
    int r = blockIdx.x * 256 + threadIdx.x;      // grid 4 x 256 -> 1024
    int c = 0;
    for (int j = 0; j < PCB; ++j) {
        if (ssrc[j] == r && c < 16) lst[r * 16 + c++] = j;
    }
    cnt[r] = c;
}

__global__ __launch_bounds__(256) void agg_build_kernel(
        const float* __restrict__ cb, const int* __restrict__ cnt,
        const int* __restrict__ lst, float2* __restrict__ agg) {
    int r = blockIdx.x, t = threadIdx.x;         // 256 threads = 256 float2 = 512 f
    float2 acc; acc.x = 0.f; acc.y = 0.f;
    int c = cnt[r];
    for (int i = 0; i < c; ++i) {
        int j = lst[r * 16 + i];
        float2 v = ((const float2*)(cb + (size_t)j * DDIM))[t];
        acc.x += v.x; acc.y += v.y;
    }
    agg[(size_t)r * 256 + t] = acc;
}

// ---------------- WMMA f32 argmin: 32 rows/block, B shared by 2 M-tiles ----------------
__global__ __launch_bounds__(256) void argmin_kernel(
        const float* __restrict__ flat, const float* __restrict__ cb,
        const float* __restrict__ cn2, int m, int* __restrict__ idx_out) {
    __shared__ float As[32][516];   // 32 rows x 512 K, padded (+4) -> conflict-free
    __shared__ float rv[8][32];
    __shared__ int   ri[8][32];

    const int t = threadIdx.x;
    const int r0 = blockIdx.x * 32;

    // stage A tile (32x512) into LDS, coalesced float4
    for (int i = 0; i < 16; ++i) {
        int e   = (i * 256 + t) * 4;
        int row = e >> 9;
        int col = e & 511;
        float4 v = *(const float4*)(flat + (size_t)(r0 + row) * DDIM + col);
        As[row][col + 0] = v.x; As[row][col + 1] = v.y;
        As[row][col + 2] = v.z; As[row][col + 3] = v.w;
    }
    __syncthreads();

    const int wave = t >> 5;
    const int lane = t & 31;
    const int lcol = lane & 15;            // A row (M) and B col (N) owned by lane
    const int koff = (lane >> 4) * 2;      // half-wave K offset per ISA layout
    const int nblk = (m + 15) >> 4;

    // slots 0-7: M-tile0 (rows r0..r0+15), slots 8-15: M-tile1 (rows r0+16..r0+31)
    float bestv[16]; int besti[16];
#pragma unroll
    for (int i = 0; i < 16; ++i) { bestv[i] = __builtin_inff(); besti[i] = 0x7FFFFFFF; }

    for (int jb = wave; jb < nblk; jb += 8) {
        const int col = (jb << 4) + lcol;
        const float* bp = cb + (size_t)col * DDIM + koff;
        v8f accA0 = {0.f,0.f,0.f,0.f,0.f,0.f,0.f,0.f};
        v8f accA1 = {0.f,0.f,0.f,0.f,0.f,0.f,0.f,0.f};
        v8f accB0 = {0.f,0.f,0.f,0.f,0.f,0.f,0.f,0.f};
        v8f accB1 = {0.f,0.f,0.f,0.f,0.f,0.f,0.f,0.f};
        for (int k = 0; k < DDIM; k += 8) {
            v2f b0  = *(const v2f*)(bp + k);
            v2f b1  = *(const v2f*)(bp + k + 4);
            v2f a00 = *(const v2f*)(&As[lcol][k + koff]);
            v2f a01 = *(const v2f*)(&As[lcol][k + 4 + koff]);
            v2f a10 = *(const v2f*)(&As[16 + lcol][k + koff]);
            v2f a11 = *(const v2f*)(&As[16 + lcol][k + 4 + koff]);
            accA0 = __builtin_amdgcn_wmma_f32_16x16x4_f32(
                        false, a00, false, b0, (short)0, accA0, false, false);
            accB0 = __builtin_amdgcn_wmma_f32_16x16x4_f32(
                        false, a10, false, b0, (short)0, accB0, false, false);
            accA1 = __builtin_amdgcn_wmma_f32_16x16x4_f32(
                        false, a01, false, b1, (short)0, accA1, false, false);
            accB1 = __builtin_amdgcn_wmma_f32_16x16x4_f32(
                        false, a11, false, b1, (short)0, accB1, false, false);
        }
        const bool valid = (col < m);
        const float cn = valid ? cn2[col] : __builtin_inff();
        const int   ci = valid ? col : 0x7FFFFFFF;
#pragma unroll
        for (int i = 0; i < 8; ++i) {
            float s0 = cn - 2.0f * (accA0[i] + accA1[i]);   // ‖c‖² − 2x·c
            float s1 = cn - 2.0f * (accB0[i] + accB1[i]);
            if (s0 < bestv[i]     || (s0 == bestv[i]     && ci < besti[i]))     { bestv[i]     = s0; besti[i]     = ci; }
            if (s1 < bestv[8 + i] || (s1 == bestv[8 + i] && ci < besti[8 + i])) { bestv[8 + i] = s1; besti[8 + i] = ci; }
        }
    }

    // reduce across the 16 lanes of each half (cols), keep first-min index
#pragma unroll
    for (int off = 8; off >= 1; off >>= 1) {
#pragma unroll
        for (int i = 0; i < 16; ++i) {
            float ov = __shfl_xor(bestv[i], off, 32);
            int   oi = __shfl_xor(besti[i], off, 32);
            if (ov < bestv[i] || (ov == bestv[i] && oi < besti[i])) {
                bestv[i] = ov; besti[i] = oi;
            }
        }
    }
    if (lcol == 0) {
        // lanes 0 / 16 hold C rows i / 8+i of each tile
        int rbase = (lane >> 4) * 8;
#pragma unroll
        for (int i = 0; i < 8; ++i) {
            rv[wave][rbase + i]      = bestv[i];     ri[wave][rbase + i]      = besti[i];
            rv[wave][16 + rbase + i] = bestv[8 + i]; ri[wave][16 + rbase + i] = besti[8 + i];
        }
    }
    __syncthreads();
    if (t < 32) {
        float bv = rv[0][t]; int bi = ri[0][t];
        for (int w = 1; w < 8; ++w) {
            float v = rv[w][t]; int ii = ri[w][t];
            if (v < bv || (v == bv && ii < bi)) { bv = v; bi = ii; }
        }
        idx_out[r0 + t] = bi;
    }
}

// ---------------- gather + STE output + deterministic partial MSE ----------------
__global__ __launch_bounds__(256) void gather_out_kernel(
        const float* __restrict__ flat, const float2* __restrict__ agg,
        const int* __restrict__ idx, float* __restrict__ out_level,
        float* __restrict__ partial) {
    __shared__ float red[256];
    const int t = threadIdx.x;
    const int tid = blockIdx.x * 256 + t;
    float s = 0.f;
    // N*D/2 = 4,194,304 float2; 524288 threads * 8
    for (int it = 0; it < 8; ++it) {
        int e  = tid + it * (GATHER_BLOCKS * 256);
        int n  = e >> 8;                 // 256 float2 per row
        int d2 = e & 255;
        float2 q = agg[(size_t)idx[n] * 256 + d2];
        float2 x = ((const float2*)flat)[e];
        float dx = q.x - x.x, dy = q.y - x.y;
        s += dx * dx + dy * dy;
        float2 o; o.x = x.x + dx; o.y = x.y + dy;   // inputs + (q - inputs)
        ((float2*)out_level)[e] = o;
    }
    red[t] = s;
    __syncthreads();
    for (int off = 128; off > 0; off >>= 1) {
        if (t < off) red[t] += red[t + off];
        __syncthreads();
    }
    if (t == 0) partial[blockIdx.x] = red[0];
}

__global__ __launch_bounds__(256) void loss_kernel(
        const float* __restrict__ partials, const float* __restrict__ prev,
        const float* __restrict__ cb, int half, float qscale, float pscale,
        float* __restrict__ loss_out) {
    __shared__ float red[256];
    const int t = threadIdx.x;
    float s = 0.f;
    for (int i = t; i < GATHER_BLOCKS; i += 256) s += partials[i];
    float p = 0.f;
    int tot = half * DDIM;
    for (int i = t; i < tot; i += 256) { float d = prev[i] - cb[i]; p += d * d; }
    red[t] = qscale * s + pscale * p;
    __syncthreads();
    for (int off = 128; off > 0; off >>= 1) {
        if (t < off) red[t] += red[t + off];
        __syncthreads();
    }
    if (t == 0) *loss_out = red[0];
}

__global__ __launch_bounds__(256) void copy_cb_kernel(
        const float* __restrict__ cb, float* __restrict__ dst) {
    int i = blockIdx.x * 256 + threadIdx.x;   // 2048 x 256 = 524288 = P*D
    dst[i] = cb[i];
}

extern "C" void kernel_launch(void* const* d_in, const int* in_sizes, int n_in,
                              void* d_out, int out_size, void* d_ws, size_t ws_size,
                              hipStream_t stream) {
    const float* inputs = (const float*)d_in[0];   // B*T*D
    const float* cb     = (const float*)d_in[1];   // P*D
    const float* prev   = (const float*)d_in[2];   // P*D
    const float* cp     = (const float*)d_in[3];   // scalar
    float* out = (float*)d_out;

    // ws carve (~150 KB)
    int*   src      = (int*)d_ws;
    int*   cnt      = src + PCB;
    int*   lst      = cnt + PCB;                   // P*16
    int*   idx      = lst + PCB * 16;              // N
    float* cn2      = (float*)(idx + NROW);        // P
    float* partials = cn2 + PCB;                   // GATHER_BLOCKS

    float*  losses = out + (size_t)NLEV * NROW * DDIM;   // 10 scalars
    // reuse the 2MB "actives" tail of d_out as agg scratch (8B-aligned), restored at end
    float2* agg    = (float2*)(losses + NLEV);

    colnorm_kernel<<<PCB, 256, 0, stream>>>(cb, cn2);

    for (int na = 0; na < NLEV; ++na) {
        const int m    = 2 << na;          // 2^(na+1)
        const int half = m >> 1;
        rng_src_kernel<<<1, 512, 0, stream>>>(cp, na, src);
        build_lists_kernel<<<4, 256, 0, stream>>>(src, cnt, lst);
        agg_build_kernel<<<PCB, 256, 0, stream>>>(cb, cnt, lst, agg);
        argmin_kernel<<<NROW / 32, 256, 0, stream>>>(inputs, cb, cn2, m, idx);
        gather_out_kernel<<<GATHER_BLOCKS, 256, 0, stream>>>(
            inputs, agg, idx, out + (size_t)na * NROW * DDIM, partials);
        const float qscale = ((na < 2) ? 1.1f : 1.0f) / ((float)NROW * (float)DDIM);
        const float pscale = ((na == 0) ? 0.0f : 0.33f) / ((float)half * (float)DDIM);
        loss_kernel<<<1, 256, 0, stream>>>(partials, prev, cb, half, qscale, pscale,
                                           losses + na);
    }
    // final actives = full codebook (overwrites agg scratch region)
    copy_cb_kernel<<<(PCB * DDIM) / 256, 256, 0, stream>>>(cb, (float*)agg);
}